// GlassBlur_43602507989290
// MI455X (gfx1250) — compile-verified
//
#include <hip/hip_runtime.h>
#include <stdint.h>

#define Hdim 1024
#define Wdim 1024
#define Cdim 3
#define RAD  2
#define TS   16
#define IN_TS (TS + 2*RAD)   // 20

typedef float v2f __attribute__((ext_vector_type(2)));
typedef float v8f __attribute__((ext_vector_type(8)));

#if defined(__has_builtin)
#  if __has_builtin(__builtin_amdgcn_wmma_f32_16x16x4_f32)
#    define HAVE_WMMA_F32 1
#  endif
#  if __has_builtin(__builtin_amdgcn_global_load_async_to_lds_b32) && \
      __has_builtin(__builtin_amdgcn_s_wait_asynccnt)
#    define HAVE_ASYNC 1
#  endif
#endif

// Normalized Gaussian taps for sigma = 0.4 (r = 2), matching float32 reference.
__device__ __forceinline__ float gk(int t) {
    switch (t) {
        case 0:  return 3.4256121e-06f;
        case 1:  return 4.0387621e-02f;
        case 2:  return 9.1921792e-01f;
        case 3:  return 4.0387621e-02f;
        default: return 3.4256121e-06f;
    }
}

// Banded horizontal-conv matrix Kh[p][n] = k[p-n] for 0 <= p-n <= 4, else 0.
__device__ __forceinline__ float hw(int p, int n) {
    int d = p - n;
    return (d >= 0 && d <= 4) ? gk(d) : 0.0f;
}

// One wave (32 lanes) produces one 16x16 output tile for all 3 channels.
// Vertical 5-tap via VALU FMA into LDS; horizontal 5-tap as five chained
// V_WMMA_F32_16X16X4_F32: Out(16x16) = Tmp(16x20) x Kh(20x16).
__global__ __launch_bounds__(32)
void blur_tile_kernel(const float* __restrict__ src, const int* __restrict__ perm,
                      float* __restrict__ dst, int clampOut) {
    __shared__ float sIn[IN_TS][IN_TS];   // 20x20 input tile (one channel)
    __shared__ float sTmp[TS][IN_TS];     // 16x20 vertically blurred

    const int lane = threadIdx.x;
    const int bx = blockIdx.x, by = blockIdx.y;
    const int gx0 = bx * TS - RAD;
    const int gy0 = by * TS - RAD;
    const int half = lane >> 4;           // 0: lanes 0-15, 1: lanes 16-31
    const int lm   = lane & 15;

    for (int c = 0; c < Cdim; ++c) {
        // ---- Stage 1: load 20x20 halo tile (edge clamp, optional perm gather)
        for (int e = lane; e < IN_TS * IN_TS; e += 32) {
            int iy = e / IN_TS, ix = e % IN_TS;
            int gy = min(max(gy0 + iy, 0), Hdim - 1);
            int gx = min(max(gx0 + ix, 0), Wdim - 1);
            int p  = gy * Wdim + gx;
            if (perm) p = perm[p];
            const float* gp = src + (size_t)p * Cdim + c;
#ifdef HAVE_ASYNC
            typedef __attribute__((address_space(1))) int* as1ip;
            typedef __attribute__((address_space(3))) int* as3ip;
            as1ip g = (as1ip)(uintptr_t)gp;
            as3ip l = (as3ip)(uint32_t)(uintptr_t)&sIn[iy][ix];
            __builtin_amdgcn_global_load_async_to_lds_b32(g, l, 0, 0);
#else
            sIn[iy][ix] = *gp;
#endif
        }
#ifdef HAVE_ASYNC
        __builtin_amdgcn_s_wait_asynccnt(0);
#endif
        __syncthreads();

        // ---- Stage 2: vertical 5-tap -> sTmp (16 rows x 20 cols)
        for (int e = lane; e < TS * IN_TS; e += 32) {
            int r = e / IN_TS, cc = e % IN_TS;
            float acc = 0.0f;
#pragma unroll
            for (int t = 0; t < 5; ++t) acc = fmaf(gk(t), sIn[r + t][cc], acc);
            sTmp[r][cc] = acc;
        }
        __syncthreads();

        // ---- Stage 3: horizontal 5-tap as WMMA chain (K split 20 -> 5x4)
        v8f acc = {};
#ifdef HAVE_WMMA_F32
#pragma unroll
        for (int j = 0; j < 5; ++j) {
            const int kb = 4 * j + half * 2;
            v2f a, b;
            // A (16x4 f32): lane holds row M=lm, K = kb, kb+1
            a.x = sTmp[lm][kb];
            a.y = sTmp[lm][kb + 1];
            // B (4x16 f32): lane holds col N=lm, rows K = kb, kb+1
            b.x = hw(kb,     lm);
            b.y = hw(kb + 1, lm);
            acc = __builtin_amdgcn_wmma_f32_16x16x4_f32(
                false, a, false, b, (short)0, acc, false, false);
        }
#else
        // Scalar fallback with identical C/D lane layout
#pragma unroll
        for (int v = 0; v < 8; ++v) {
            int m = v + half * 8;
            float s = 0.0f;
#pragma unroll
            for (int t = 0; t < 5; ++t) s = fmaf(gk(t), sTmp[m][lm + t], s);
            acc[v] = s;
        }
#endif

        // ---- Stage 4: store D (VGPR v: M = v + 8*half, N = lm)
#pragma unroll
        for (int v = 0; v < 8; ++v) {
            int row = by * TS + v + half * 8;
            int col = bx * TS + lm;
            float val = acc[v];
            if (clampOut) val = fminf(fmaxf(val, 0.0f), 1.0f);
            dst[((size_t)row * Wdim + col) * Cdim + c] = val;
        }
        __syncthreads();   // sIn reused next channel
    }
}

__global__ void init_perm_kernel(int* perm, int n) {
    int i = blockIdx.x * blockDim.x + threadIdx.x;
    if (i < n) perm[i] = i;
}

// Skewed-wavefront replay of the sequential swap chain.
// Thread r owns row hh = 1023 - r; at step t it executes column index
// c = t - 2r (ww = 1023 - c). Row r-1 (earlier in sequence) stays >= 2
// columns ahead, which is exactly the dependence distance of the swaps.
__global__ __launch_bounds__(1024)
void perm_swap_kernel(const int* __restrict__ swaps, int* perm) {
    const int r     = threadIdx.x;
    const int nrows = Hdim - 2;                 // 1022
    const int ncols = Wdim - 2;                 // 1022
    const int steps = ncols + 2 * (nrows - 1);  // makespan of the wavefront

    for (int t = 0; t < steps; ++t) {
        int cidx = t - 2 * r;
        if (r < nrows && cidx >= 0 && cidx < ncols) {
            int hh = (Hdim - 1) - r;            // 1023 .. 2
            int ww = (Wdim - 1) - cidx;         // 1023 .. 2
            int sidx = r * ncols + cidx;
            int dx = swaps[2 * sidx + 0];       // in {-1, 0}
            int dy = swaps[2 * sidx + 1];
            int a = hh * Wdim + ww;
            int b = (hh + dy) * Wdim + (ww + dx);
            if (a != b) {
                int pa = perm[a];
                int pb = perm[b];
                perm[a] = pb;
                perm[b] = pa;
            }
        }
        __threadfence();   // make swaps visible across the 32 waves (one WGP)
        __syncthreads();
    }
}

extern "C" void kernel_launch(void* const* d_in, const int* in_sizes, int n_in,
                              void* d_out, int out_size, void* d_ws, size_t ws_size,
                              hipStream_t stream) {
    (void)in_sizes; (void)n_in; (void)out_size; (void)ws_size;
    const float* img   = (const float*)d_in[0];
    const int*   swaps = (const int*)d_in[1];
    float*       out   = (float*)d_out;

    float* tmp1 = (float*)d_ws;                                    // 12 MB
    int*   perm = (int*)((char*)d_ws +
                         (size_t)Hdim * Wdim * Cdim * sizeof(float)); // +4 MB

    init_perm_kernel<<<(Hdim * Wdim + 255) / 256, 256, 0, stream>>>(perm, Hdim * Wdim);
    blur_tile_kernel<<<dim3(Wdim / TS, Hdim / TS), 32, 0, stream>>>(img, nullptr, tmp1, 0);
    perm_swap_kernel<<<1, 1024, 0, stream>>>(swaps, perm);
    blur_tile_kernel<<<dim3(Wdim / TS, Hdim / TS), 32, 0, stream>>>(tmp1, perm, out, 1);
}